// NTM_43233140801818
// MI455X (gfx1250) — compile-verified
//
#include <hip/hip_runtime.h>
#include <math.h>

typedef __attribute__((ext_vector_type(2))) float v2f;
typedef __attribute__((ext_vector_type(8))) float v8f;
typedef __attribute__((ext_vector_type(4))) unsigned int u32x4;
typedef __attribute__((ext_vector_type(8))) int i32x8;
typedef __attribute__((ext_vector_type(4))) int i32x4;

#define EPS 1e-8f

__device__ __forceinline__ v8f wmma4(v2f a, v2f b, v8f c) {
    // D = A(16x4) * B(4x16) + C(16x16), f32
    return __builtin_amdgcn_wmma_f32_16x16x4_f32(
        false, a, false, b, (short)0, c, false, false);
}

// ---------------- TDM: 2D tile load HBM -> LDS ----------------------------
// D# per CDNA5 ISA. data_size=2 (4B elements).
// pad_interval code c -> 2^(c+1) DWORDs; pad_amount code p -> p+1 DWORDs.
__device__ __forceinline__ void tdm_load_2d(
    unsigned lds_off, const void* gptr,
    unsigned tensor_d0, unsigned tensor_d1,
    unsigned tile_d0, unsigned tile_d1, unsigned stride0,
    unsigned pad_int_code, unsigned pad_amt_code) {
    unsigned long long ga = (unsigned long long)gptr;
    u32x4 g0;
    g0[0] = 1u;  // count=1, user descriptor, no gather
    g0[1] = lds_off;
    g0[2] = (unsigned)(ga & 0xFFFFFFFFu);
    g0[3] = (unsigned)((ga >> 32) & 0x01FFFFFFu) | (2u << 30);  // type=2 (image)
    const unsigned pe = (pad_amt_code | pad_int_code) ? 1u : 0u;
    i32x8 g1;
    g1[0] = (int)((2u << 16) | (pe << 20) | (pad_int_code << 22) | (pad_amt_code << 25));
    g1[1] = (int)((tensor_d0 & 0xFFFFu) << 16);
    g1[2] = (int)(((tensor_d0 >> 16) & 0xFFFFu) | ((tensor_d1 & 0xFFFFu) << 16));
    g1[3] = (int)(((tensor_d1 >> 16) & 0xFFFFu) | ((tile_d0 & 0xFFFFu) << 16));
    g1[4] = (int)(tile_d1 & 0xFFFFu);  // tile_dim2 = 0
    g1[5] = (int)stride0;              // tensor_dim0_stride low 32
    g1[6] = 0;
    g1[7] = 0;
    i32x4 z4 = {0, 0, 0, 0};
#if defined(__clang_major__) && (__clang_major__ >= 23)
    i32x8 z8 = {0, 0, 0, 0, 0, 0, 0, 0};
    __builtin_amdgcn_tensor_load_to_lds(g0, g1, z4, z4, z8, 0);
#else
    __builtin_amdgcn_tensor_load_to_lds(g0, g1, z4, z4, 0);
#endif
}

__device__ __forceinline__ unsigned lds_offset_of(const void* p) {
    return (unsigned)(unsigned long long)p;  // LDS aperture: low 32 bits = LDS addr
}

// ---------------- K1: controller MLP (tiny) -------------------------------
__global__ __launch_bounds__(256) void ctrl_kernel(
    const float* __restrict__ prog, const float* __restrict__ W_ctrl,
    const float* __restrict__ b_ctrl, const float* __restrict__ W_beta,
    const float* __restrict__ b_beta, float* __restrict__ h_out,
    float* __restrict__ beta_out) {
    __shared__ float red[256];
    const int d = threadIdx.x;
    float hv[8];
#pragma unroll
    for (int t = 0; t < 8; ++t) {
        float acc = b_ctrl[d];
#pragma unroll
        for (int f = 0; f < 32; ++f) acc += prog[t * 32 + f] * W_ctrl[f * 256 + d];
        hv[t] = fmaxf(acc, 0.0f);
        h_out[t * 256 + d] = hv[t];
    }
    const float wb = W_beta[d];
    for (int t = 0; t < 8; ++t) {
        red[d] = hv[t] * wb;
        __syncthreads();
        for (int s = 128; s > 0; s >>= 1) {
            if (d < s) red[d] += red[d + s];
            __syncthreads();
        }
        if (d == 0) {
            float x = red[0] + b_beta[0];
            beta_out[t] = (x > 20.0f) ? x : log1pf(expf(x));  // softplus
        }
        __syncthreads();
    }
}

// ---------------- K2: read keys k = tanh(h @ W_key + b_key) ---------------
__global__ __launch_bounds__(256) void keys_kernel(
    const float* __restrict__ h, const float* __restrict__ W_key,
    const float* __restrict__ b_key, float* __restrict__ k) {
    __shared__ float hs[8 * 256];
    const int tid = threadIdx.x;
    for (int i = tid; i < 2048; i += 256) hs[i] = h[i];
    __syncthreads();
    const int n = blockIdx.x * 256 + tid;
    float acc[8] = {0, 0, 0, 0, 0, 0, 0, 0};
    for (int d = 0; d < 256; ++d) {
        const float wv = W_key[(size_t)d * 4096 + n];
#pragma unroll
        for (int t = 0; t < 8; ++t) acc[t] += hs[t * 256 + d] * wv;
    }
    const float bk = b_key[n];
#pragma unroll
    for (int t = 0; t < 8; ++t) k[t * 4096 + n] = tanhf(acc[t] + bk);
#pragma unroll
    for (int t = 8; t < 16; ++t) k[t * 4096 + n] = 0.0f;
}

// ---------------- K3: fold beta/||k|| into k ------------------------------
__global__ __launch_bounds__(256) void scale_kernel(
    const float* __restrict__ beta, float* __restrict__ k) {
    __shared__ float red[256];
    __shared__ float sc[8];
    const int tid = threadIdx.x;
    for (int t = 0; t < 8; ++t) {
        float ss = 0.0f;
        for (int n = tid; n < 4096; n += 256) {
            float v = k[t * 4096 + n];
            ss += v * v;
        }
        red[tid] = ss;
        __syncthreads();
        for (int s = 128; s > 0; s >>= 1) {
            if (tid < s) red[tid] += red[tid + s];
            __syncthreads();
        }
        if (tid == 0) sc[t] = beta[t] / (sqrtf(red[0]) + EPS);
        __syncthreads();
    }
#pragma unroll
    for (int t = 0; t < 8; ++t) {
        const float s = sc[t];
        for (int n = tid; n < 4096; n += 256) k[t * 4096 + n] *= s;
    }
}

// ---------------- K4: scores pass, TDM double-buffered --------------------
// S[t][j] = dot(scaled_k_t, mem_j) / (||mem_j|| + eps). One HBM pass.
#define SC_PITCH 130  // TDM pad: 2 DW every 128 DW (codes 6,1) -> bank stride 2
__global__ __launch_bounds__(256) void scores_kernel(
    const float* __restrict__ mem, const float* __restrict__ k,
    float* __restrict__ S) {
    __shared__ float tile[2][128 * SC_PITCH];
    __shared__ float ktile[16 * SC_PITCH];
    __shared__ float norm2[128];
    const int tid = threadIdx.x;
    const int lane = tid & 31;
    const int wave = tid >> 5;  // 8 waves x 16 rows
    const int jbase = blockIdx.x * 128;
    const int arow = lane & 15;
    const int asub = (lane >> 4) * 2;

    if (tid < 128) norm2[tid] = 0.0f;

    // prologue: async-load chunk 0 into buffer 0 (one TDM issue, wave 0)
    if (wave == 0) {
        tdm_load_2d(lds_offset_of(&tile[0][0]),
                    mem + (size_t)jbase * 4096, 4096, 16384, 128, 128, 4096, 6, 1);
    }

    v8f c = {0.f, 0.f, 0.f, 0.f, 0.f, 0.f, 0.f, 0.f};

    for (int i = 0; i < 32; ++i) {
        const int cur = i & 1;
        const int kb = i * 128;
        // cooperative load of 16x128 key tile for this chunk
        for (int r = 0; r < 2; ++r) {
            const int q = tid + 256 * r;  // 0..511 float4s
            const int row = q >> 5, c4 = q & 31;
            const float4 v = *(const float4*)(k + (size_t)row * 4096 + kb + 4 * c4);
            float* dst = &ktile[row * SC_PITCH + 4 * c4];
            v2f lo = {v.x, v.y}, hi = {v.z, v.w};
            *(v2f*)dst = lo;
            *(v2f*)(dst + 2) = hi;
        }
        if (wave == 0) __builtin_amdgcn_s_wait_tensorcnt(0);
        __syncthreads();  // chunk i resident in tile[cur], ktile ready
        // prefetch chunk i+1 into the other buffer while we compute
        if (wave == 0 && i + 1 < 32) {
            tdm_load_2d(lds_offset_of(&tile[cur ^ 1][0]),
                        mem + (size_t)jbase * 4096 + kb + 128,
                        4096, 16384, 128, 128, 4096, 6, 1);
        }
        // row-norm partials from LDS (2 threads per row)
        {
            const int row = tid >> 1;
            const float* p = &tile[cur][row * SC_PITCH + (tid & 1) * 64];
            float s = 0.0f;
#pragma unroll
            for (int x = 0; x < 64; x += 2) {
                v2f v = *(const v2f*)(p + x);
                s += v.x * v.x + v.y * v.y;
            }
            atomicAdd(&norm2[row], s);
        }
        // 32 WMMAs on this chunk per wave
#pragma unroll 8
        for (int kc = 0; kc < 128; kc += 4) {
            v2f a = *(const v2f*)&ktile[arow * SC_PITCH + kc + asub];
            v2f b = *(const v2f*)&tile[cur][(wave * 16 + arow) * SC_PITCH + kc + asub];
            c = wmma4(a, b, c);
        }
        __syncthreads();  // done with tile[cur] before TDM refills it
    }
    if (lane < 16) {
        const float nrm = sqrtf(norm2[wave * 16 + lane]) + EPS;
        const int j = jbase + wave * 16 + lane;
#pragma unroll
        for (int v = 0; v < 8; ++v) S[v * 16384 + j] = c[v] / nrm;
    }
}

// ---------------- K5: softmax over (8, 16384), in place -------------------
__global__ __launch_bounds__(256) void softmax_kernel(float* __restrict__ S) {
    __shared__ float red[256];
    float* row = S + blockIdx.x * 16384;
    const int tid = threadIdx.x;
    float m = -INFINITY;
    for (int i = tid; i < 16384; i += 256) m = fmaxf(m, row[i]);
    red[tid] = m;
    __syncthreads();
    for (int s = 128; s > 0; s >>= 1) {
        if (tid < s) red[tid] = fmaxf(red[tid], red[tid + s]);
        __syncthreads();
    }
    m = red[0];
    __syncthreads();
    float sum = 0.0f;
    for (int i = tid; i < 16384; i += 256) sum += expf(row[i] - m);
    red[tid] = sum;
    __syncthreads();
    for (int s = 128; s > 0; s >>= 1) {
        if (tid < s) red[tid] += red[tid + s];
        __syncthreads();
    }
    const float inv = 1.0f / red[0];
    for (int i = tid; i < 16384; i += 256) row[i] = expf(row[i] - m) * inv;
}

// ---------------- K6: reads partials = w @ memory, TDM double-buffered ----
// grid (32 k-slices x 8 n-slices). TDM pad: 4 DW every 256 DW (codes 7,3).
// element (r,c) of a 32x512 tile lives at r*520 + c + 4*(c>>8).
#define RD_PITCH 520
__global__ __launch_bounds__(256) void reads_partial_kernel(
    const float* __restrict__ mem, const float* __restrict__ wgt,
    float* __restrict__ part) {
    __shared__ float tile[2][32 * RD_PITCH];
    const int tid = threadIdx.x;
    const int lane = tid & 31;
    const int wave = tid >> 5;
    const int k0 = blockIdx.x * 512;
    const int n0 = blockIdx.y * 512;
    const int arow = lane & 15;
    const int asub = (lane >> 4) * 2;
    const int wn = wave * 64;
    int coladj[4];
#pragma unroll
    for (int t4 = 0; t4 < 4; ++t4) {
        const int col = wn + t4 * 16 + arow;
        coladj[t4] = col + ((col >> 8) << 2);
    }

    if (wave == 0) {
        tdm_load_2d(lds_offset_of(&tile[0][0]),
                    mem + (size_t)k0 * 4096 + n0, 4096, 16384, 512, 32, 4096, 7, 3);
    }

    v8f acc[4];
#pragma unroll
    for (int t4 = 0; t4 < 4; ++t4) acc[t4] = (v8f){0.f, 0.f, 0.f, 0.f, 0.f, 0.f, 0.f, 0.f};

    for (int i = 0; i < 16; ++i) {  // 16 chunks of 32 rows
        const int cur = i & 1;
        const int rc = i * 32;
        if (wave == 0) __builtin_amdgcn_s_wait_tensorcnt(0);
        __syncthreads();
        if (wave == 0 && i + 1 < 16) {
            tdm_load_2d(lds_offset_of(&tile[cur ^ 1][0]),
                        mem + (size_t)(k0 + rc + 32) * 4096 + n0,
                        4096, 16384, 512, 32, 4096, 7, 3);
        }
#pragma unroll
        for (int kc = 0; kc < 32; kc += 4) {
            const int kg = k0 + rc + kc + asub;  // global memory-row for A frag
            v2f a;
            a.x = (arow < 8) ? wgt[arow * 16384 + kg] : 0.0f;
            a.y = (arow < 8) ? wgt[arow * 16384 + kg + 1] : 0.0f;
            const int krow = (kc + asub) * RD_PITCH;
#pragma unroll
            for (int t4 = 0; t4 < 4; ++t4) {
                v2f b;
                b.x = tile[cur][krow + coladj[t4]];
                b.y = tile[cur][krow + RD_PITCH + coladj[t4]];
                acc[t4] = wmma4(a, b, acc[t4]);
            }
        }
        __syncthreads();
    }
    if (lane < 16) {
#pragma unroll
        for (int t4 = 0; t4 < 4; ++t4)
#pragma unroll
            for (int v = 0; v < 8; ++v)
                part[((size_t)blockIdx.x * 8 + v) * 4096 + n0 + wn + t4 * 16 + lane] =
                    acc[t4][v];
    }
}

// ---------------- K7: reduce split-K partials -> reads(8x4096) ------------
__global__ __launch_bounds__(256) void reads_reduce_kernel(
    const float* __restrict__ part, float* __restrict__ reads) {
    const int i = blockIdx.x * 256 + threadIdx.x;
    float s = 0.0f;
    for (int kb = 0; kb < 32; ++kb) s += part[(size_t)kb * 8 * 4096 + i];
    reads[i] = s;
}

// ---------------- K8: executioner, fully fused ----------------------------
#define XT_PITCH 514
#define CB_PITCH 66
__global__ __launch_bounds__(128) void exec_kernel(
    const float* __restrict__ X, const float* __restrict__ iemb,
    const float* __restrict__ reads, const float* __restrict__ oemb,
    float* __restrict__ out) {
    __shared__ float xt[64 * XT_PITCH];
    __shared__ float cb[4 * 16 * CB_PITCH];
    const int tid = threadIdx.x;
    const int lane = tid & 31;
    const int wave = tid >> 5;
    const int r0 = blockIdx.x * 64;
    const int arow = lane & 15;
    const int asub = (lane >> 4) * 2;
    const int rhalf = (lane < 16) ? 0 : 8;
    float* mycb = &cb[wave * 16 * CB_PITCH];

    for (int r = 0; r < 64; ++r) {
        const int q = tid + 128 * r;
        const int row = q >> 7, c4 = q & 127;
        const float4 v = *(const float4*)(X + (size_t)(r0 + row) * 512 + 4 * c4);
        float* dst = &xt[row * XT_PITCH + 4 * c4];
        v2f lo = {v.x, v.y}, hi = {v.z, v.w};
        *(v2f*)dst = lo;
        *(v2f*)(dst + 2) = hi;
    }
    __syncthreads();

    v8f acc[4];
#pragma unroll
    for (int t4 = 0; t4 < 4; ++t4) acc[t4] = (v8f){0.f, 0.f, 0.f, 0.f, 0.f, 0.f, 0.f, 0.f};
    for (int kc = 0; kc < 512; kc += 4) {
        v2f a = *(const v2f*)&xt[(wave * 16 + arow) * XT_PITCH + kc + asub];
#pragma unroll
        for (int t4 = 0; t4 < 4; ++t4) {
            v2f b;
            b.x = iemb[(kc + asub) * 64 + t4 * 16 + arow];
            b.y = iemb[(kc + asub + 1) * 64 + t4 * 16 + arow];
            acc[t4] = wmma4(a, b, acc[t4]);
        }
    }
#pragma unroll
    for (int t4 = 0; t4 < 4; ++t4)
#pragma unroll
        for (int v = 0; v < 8; ++v)
            mycb[(v + rhalf) * CB_PITCH + t4 * 16 + arow] = acc[t4][v];

    for (int t = 0; t < 8; ++t) {
        const float* Wt = reads + t * 4096;
        v8f nacc[4];
#pragma unroll
        for (int t4 = 0; t4 < 4; ++t4)
            nacc[t4] = (v8f){0.f, 0.f, 0.f, 0.f, 0.f, 0.f, 0.f, 0.f};
#pragma unroll
        for (int kc = 0; kc < 64; kc += 4) {
            v2f a = *(const v2f*)&mycb[arow * CB_PITCH + kc + asub];
#pragma unroll
            for (int t4 = 0; t4 < 4; ++t4) {
                v2f b;
                b.x = Wt[(kc + asub) * 64 + t4 * 16 + arow];
                b.y = Wt[(kc + asub + 1) * 64 + t4 * 16 + arow];
                nacc[t4] = wmma4(a, b, nacc[t4]);
            }
        }
#pragma unroll
        for (int t4 = 0; t4 < 4; ++t4)
#pragma unroll
            for (int v = 0; v < 8; ++v)
                mycb[(v + rhalf) * CB_PITCH + t4 * 16 + arow] = nacc[t4][v];
    }

    for (int g = 0; g < 8; ++g) {
        v8f oacc[4];
#pragma unroll
        for (int t4 = 0; t4 < 4; ++t4)
            oacc[t4] = (v8f){0.f, 0.f, 0.f, 0.f, 0.f, 0.f, 0.f, 0.f};
#pragma unroll
        for (int kc = 0; kc < 64; kc += 4) {
            v2f a = *(const v2f*)&mycb[arow * CB_PITCH + kc + asub];
#pragma unroll
            for (int t4 = 0; t4 < 4; ++t4) {
                v2f b;
                b.x = oemb[(kc + asub) * 512 + g * 64 + t4 * 16 + arow];
                b.y = oemb[(kc + asub + 1) * 512 + g * 64 + t4 * 16 + arow];
                oacc[t4] = wmma4(a, b, oacc[t4]);
            }
        }
        const int rbase = r0 + wave * 16 + rhalf;
#pragma unroll
        for (int t4 = 0; t4 < 4; ++t4)
#pragma unroll
            for (int v = 0; v < 8; ++v)
                out[(size_t)(rbase + v) * 512 + g * 64 + t4 * 16 + arow] = oacc[t4][v];
    }
}

// ---------------- launcher ------------------------------------------------
extern "C" void kernel_launch(void* const* d_in, const int* in_sizes, int n_in,
                              void* d_out, int out_size, void* d_ws, size_t ws_size,
                              hipStream_t stream) {
    (void)in_sizes; (void)n_in; (void)out_size; (void)ws_size;
    const float* X       = (const float*)d_in[0];
    const float* program = (const float*)d_in[1];
    const float* memory  = (const float*)d_in[2];
    const float* iemb    = (const float*)d_in[3];
    const float* oemb    = (const float*)d_in[4];
    const float* W_ctrl  = (const float*)d_in[5];
    const float* b_ctrl  = (const float*)d_in[6];
    const float* W_key   = (const float*)d_in[7];
    const float* b_key   = (const float*)d_in[8];
    const float* W_beta  = (const float*)d_in[9];
    const float* b_beta  = (const float*)d_in[10];

    float* ws    = (float*)d_ws;
    float* h     = ws;                    // 8*256
    float* beta  = ws + 2048;             // 8 (pad 32)
    float* k     = ws + 2080;             // 16*4096 (padded key rows)
    float* S     = k + 65536;             // 8*16384 (scores -> softmax in place)
    float* part  = S + 131072;            // 32*8*4096 split-K partials
    float* reads = part + 32 * 8 * 4096;  // 8*4096
    float* out   = (float*)d_out;

    ctrl_kernel<<<1, 256, 0, stream>>>(program, W_ctrl, b_ctrl, W_beta, b_beta, h, beta);
    keys_kernel<<<16, 256, 0, stream>>>(h, W_key, b_key, k);
    scale_kernel<<<1, 256, 0, stream>>>(beta, k);
    scores_kernel<<<128, 256, 0, stream>>>(memory, k, S);
    softmax_kernel<<<8, 256, 0, stream>>>(S);
    reads_partial_kernel<<<dim3(32, 8), 256, 0, stream>>>(memory, S, part);
    reads_reduce_kernel<<<128, 256, 0, stream>>>(part, reads);
    exec_kernel<<<64, 128, 0, stream>>>(X, iemb, reads, oemb, out);
}